// Dnafromer_2121713844582
// MI455X (gfx1250) — compile-verified
//
#include <hip/hip_runtime.h>
#include <math.h>

typedef __attribute__((ext_vector_type(2))) float v2f;
typedef __attribute__((ext_vector_type(8))) float v8f;

#define SEQ   100
#define DMODEL 4
#define BATCH 256
#define ROWS  (BATCH * SEQ)   // 25600
#define FFDIM 2048
#define NREP  8

__device__ __forceinline__ float gelu_exact(float x) {
    return 0.5f * x * (1.0f + erff(x * 0.70710678118654752440f));
}

// ---------------------------------------------------------------------------
// K1: h = x*cw+cb ; 8x { gelu(LN_[100x4](h@Wl^T+bl)) } ; h = h@Wl^T+bl
// one block per batch element, thread r owns row r (r<100)
// ---------------------------------------------------------------------------
__global__ void k_pre(const float* __restrict__ x, const float* __restrict__ conv_w,
                      const float* __restrict__ conv_b,
                      const float* __restrict__ Wl, const float* __restrict__ bl,
                      const float* __restrict__ lng, const float* __restrict__ lnb,
                      float* __restrict__ h) {
    __shared__ float s1[128];
    __shared__ float s2[128];
    const int b = blockIdx.x;
    const int r = threadIdx.x;
    const bool act = (r < SEQ);

    const float cw = conv_w[0], cb = conv_b[0];
    float hv[4] = {0.f, 0.f, 0.f, 0.f};
    if (act) {
        const float* xp = x + (b * SEQ + r) * DMODEL;
        #pragma unroll
        for (int c = 0; c < 4; ++c) hv[c] = xp[c] * cw + cb;
    }
    float W[16], bb[4], g[4] = {1,1,1,1}, be[4] = {0,0,0,0};
    #pragma unroll
    for (int j = 0; j < 16; ++j) W[j] = Wl[j];
    #pragma unroll
    for (int c = 0; c < 4; ++c) bb[c] = bl[c];
    if (act) {
        #pragma unroll
        for (int c = 0; c < 4; ++c) { g[c] = lng[r*4 + c]; be[c] = lnb[r*4 + c]; }
    }

    for (int it = 0; it < NREP; ++it) {
        float y[4];
        #pragma unroll
        for (int c = 0; c < 4; ++c) {
            float a = bb[c];
            #pragma unroll
            for (int k = 0; k < 4; ++k) a += hv[k] * W[c*4 + k];
            y[c] = a;
        }
        float ls = 0.f, lq = 0.f;
        if (act) {
            #pragma unroll
            for (int c = 0; c < 4; ++c) { ls += y[c]; lq += y[c]*y[c]; }
        }
        s1[r] = ls; s2[r] = lq;
        __syncthreads();
        for (int off = 64; off; off >>= 1) {
            if (r < off) { s1[r] += s1[r + off]; s2[r] += s2[r + off]; }
            __syncthreads();
        }
        const float mean = s1[0] * (1.0f / 400.0f);
        const float var  = s2[0] * (1.0f / 400.0f) - mean * mean;
        const float inv  = rsqrtf(var + 1e-5f);
        __syncthreads();
        if (act) {
            #pragma unroll
            for (int c = 0; c < 4; ++c)
                hv[c] = gelu_exact((y[c] - mean) * inv * g[c] + be[c]);
        }
    }
    if (act) {
        float* hp = h + (b * SEQ + r) * DMODEL;
        #pragma unroll
        for (int c = 0; c < 4; ++c) {
            float a = bb[c];
            #pragma unroll
            for (int k = 0; k < 4; ++k) a += hv[k] * W[c*4 + k];
            hp[c] = a;
        }
    }
}

// ---------------------------------------------------------------------------
// K2: qkv = h @ Win^T + bi  (per-row 4 -> 12)
// ---------------------------------------------------------------------------
__global__ void k_qkv(const float* __restrict__ h, const float* __restrict__ Win,
                      const float* __restrict__ bi, float* __restrict__ qkv) {
    const int row = blockIdx.x * blockDim.x + threadIdx.x;
    if (row >= ROWS) return;
    const float* hp = h + row * 4;
    float hv[4];
    #pragma unroll
    for (int k = 0; k < 4; ++k) hv[k] = hp[k];
    float* qp = qkv + row * 12;
    #pragma unroll
    for (int j = 0; j < 12; ++j) {
        float a = bi[j];
        #pragma unroll
        for (int k = 0; k < 4; ++k) a += hv[k] * Win[j*4 + k];
        qp[j] = a;
    }
}

// ---------------------------------------------------------------------------
// K3: rank-1 attention over the S=256 axis, head_dim=1.
// block = (position n, head hh); thread s streams 256 k/v from LDS.
// exact row max via kmin/kmax since score(s,t) = q_s * k_t.
// ---------------------------------------------------------------------------
__global__ void k_attn(const float* __restrict__ qkv, float* __restrict__ o) {
    __shared__ float sK[256];
    __shared__ float sV[256];
    __shared__ float sR[256];
    const int n  = blockIdx.x >> 2;
    const int hh = blockIdx.x & 3;
    const int s  = threadIdx.x;
    const float* base = qkv + (s * SEQ + n) * 12;
    const float q = base[hh];
    sK[s] = base[4 + hh];
    sV[s] = base[8 + hh];
    __syncthreads();
    sR[s] = sK[s]; __syncthreads();
    for (int off = 128; off; off >>= 1) {
        if (s < off) sR[s] = fmaxf(sR[s], sR[s + off]);
        __syncthreads();
    }
    const float kmax = sR[0]; __syncthreads();
    sR[s] = sK[s]; __syncthreads();
    for (int off = 128; off; off >>= 1) {
        if (s < off) sR[s] = fminf(sR[s], sR[s + off]);
        __syncthreads();
    }
    const float kmin = sR[0];
    const float m = q * ((q >= 0.f) ? kmax : kmin);
    float den = 0.f, num = 0.f;
    for (int t = 0; t < 256; ++t) {
        const float e = __expf(q * sK[t] - m);
        den += e;
        num += e * sV[t];
    }
    o[(s * SEQ + n) * 4 + hh] = num / den;
}

// ---------------------------------------------------------------------------
// K4: h = LN_row( h + o@Wout^T + bout )
// ---------------------------------------------------------------------------
__global__ void k_proj_ln(const float* __restrict__ o, const float* __restrict__ Wout,
                          const float* __restrict__ bout, const float* __restrict__ g1,
                          const float* __restrict__ be1, float* __restrict__ h) {
    const int row = blockIdx.x * blockDim.x + threadIdx.x;
    if (row >= ROWS) return;
    const float* op = o + row * 4;
    float* hp = h + row * 4;
    float ov[4], y[4];
    #pragma unroll
    for (int k = 0; k < 4; ++k) ov[k] = op[k];
    #pragma unroll
    for (int c = 0; c < 4; ++c) {
        float a = bout[c];
        #pragma unroll
        for (int k = 0; k < 4; ++k) a += ov[k] * Wout[c*4 + k];
        y[c] = hp[c] + a;
    }
    const float mean = 0.25f * (y[0] + y[1] + y[2] + y[3]);
    float var = 0.f;
    #pragma unroll
    for (int c = 0; c < 4; ++c) { const float d0 = y[c] - mean; var += d0 * d0; }
    var *= 0.25f;
    const float inv = rsqrtf(var + 1e-5f);
    #pragma unroll
    for (int c = 0; c < 4; ++c) hp[c] = (y[c] - mean) * inv * g1[c] + be1[c];
}

// ---------------------------------------------------------------------------
// K5: fused FF block on the matrix pipe.
// One wave per 16-row tile. GEMM1 (K=4) = v_wmma_f32_16x16x4_f32 per 16 FF
// columns; bias+relu on the D tile; GEMM2 (2048 -> 4) accumulated per lane
// straight from the WMMA D layout, cross-lane reduced once at the end.
// Then +b2, residual, per-row LN. In-place on h (tiles are disjoint).
// ---------------------------------------------------------------------------
__global__ void __launch_bounds__(128) k_ffn(const float* __restrict__ h,
                      const float* __restrict__ W1, const float* __restrict__ b1,
                      const float* __restrict__ W2, const float* __restrict__ b2,
                      const float* __restrict__ g2, const float* __restrict__ be2,
                      float* __restrict__ hout) {
    const int wave  = threadIdx.x >> 5;
    const int lane  = threadIdx.x & 31;
    const int tile  = blockIdx.x * 4 + wave;
    const int rbase = tile * 16;
    const int lm    = lane & 15;
    const int half  = lane >> 4;

    // A-layout (16x4 f32): lane = M%16, VGPR0/1 = K {0,1} (lanes 0-15) / {2,3} (16-31)
    const float* ap = h + (rbase + lm) * 4 + half * 2;
    v2f a; a.x = ap[0]; a.y = ap[1];
    const v8f cz = {0.f, 0.f, 0.f, 0.f, 0.f, 0.f, 0.f, 0.f};

    float zacc[8][4];
    #pragma unroll
    for (int r2 = 0; r2 < 8; ++r2)
        #pragma unroll
        for (int c = 0; c < 4; ++c) zacc[r2][c] = 0.f;

    for (int ch = 0; ch < FFDIM / 16; ++ch) {
        const int ffl = ch * 16 + lm;
        // B-layout (4x16 f32): lane = N%16, VGPR0/1 = K {0,1}/{2,3}; W1 is [2048,4] row-major
        const float* bp = W1 + ffl * 4 + half * 2;
        v2f bm; bm.x = bp[0]; bm.y = bp[1];

        v8f d = __builtin_amdgcn_wmma_f32_16x16x4_f32(
            false, a, false, bm, (short)0, cz, false, false);

        const float b1v = b1[ffl];
        const float w20 = W2[0 * FFDIM + ffl];
        const float w21 = W2[1 * FFDIM + ffl];
        const float w22 = W2[2 * FFDIM + ffl];
        const float w23 = W2[3 * FFDIM + ffl];
        #pragma unroll
        for (int r2 = 0; r2 < 8; ++r2) {
            // D layout: VGPR r2 holds row M = r2 + 8*half, column N = ffl
            float f = d[r2] + b1v;
            f = fmaxf(f, 0.f);
            zacc[r2][0] += f * w20;
            zacc[r2][1] += f * w21;
            zacc[r2][2] += f * w22;
            zacc[r2][3] += f * w23;
        }
    }

    // reduce across the 16 lanes of each half (masks < 16 keep halves separate)
    #pragma unroll
    for (int r2 = 0; r2 < 8; ++r2)
        #pragma unroll
        for (int c = 0; c < 4; ++c) {
            float v = zacc[r2][c];
            v += __shfl_xor(v, 1, 32);
            v += __shfl_xor(v, 2, 32);
            v += __shfl_xor(v, 4, 32);
            v += __shfl_xor(v, 8, 32);
            zacc[r2][c] = v;
        }

    if (lm == 0) {  // lanes 0 and 16 own rows rbase+0..7 and rbase+8..15
        #pragma unroll
        for (int r2 = 0; r2 < 8; ++r2) {
            const int row = rbase + half * 8 + r2;
            const float* hp = h + row * 4;
            float y[4];
            #pragma unroll
            for (int c = 0; c < 4; ++c) y[c] = hp[c] + zacc[r2][c] + b2[c];
            const float mean = 0.25f * (y[0] + y[1] + y[2] + y[3]);
            float var = 0.f;
            #pragma unroll
            for (int c = 0; c < 4; ++c) { const float d0 = y[c] - mean; var += d0 * d0; }
            var *= 0.25f;
            const float inv = rsqrtf(var + 1e-5f);
            float* op = hout + row * 4;
            #pragma unroll
            for (int c = 0; c < 4; ++c) op[c] = (y[c] - mean) * inv * g2[c] + be2[c];
        }
    }
}

// ---------------------------------------------------------------------------
// K6: out = softmax_D( h @ Wl^T + bl )
// ---------------------------------------------------------------------------
__global__ void k_final(const float* __restrict__ h, const float* __restrict__ Wl,
                        const float* __restrict__ bl, float* __restrict__ out) {
    const int row = blockIdx.x * blockDim.x + threadIdx.x;
    if (row >= ROWS) return;
    const float* hp = h + row * 4;
    float hv[4], y[4];
    #pragma unroll
    for (int k = 0; k < 4; ++k) hv[k] = hp[k];
    #pragma unroll
    for (int c = 0; c < 4; ++c) {
        float a = bl[c];
        #pragma unroll
        for (int k = 0; k < 4; ++k) a += hv[k] * Wl[c*4 + k];
        y[c] = a;
    }
    const float m = fmaxf(fmaxf(y[0], y[1]), fmaxf(y[2], y[3]));
    float e[4], s = 0.f;
    #pragma unroll
    for (int c = 0; c < 4; ++c) { e[c] = __expf(y[c] - m); s += e[c]; }
    const float invs = 1.f / s;
    float* op = out + row * 4;
    #pragma unroll
    for (int c = 0; c < 4; ++c) op[c] = e[c] * invs;
}

// ---------------------------------------------------------------------------
extern "C" void kernel_launch(void* const* d_in, const int* in_sizes, int n_in,
                              void* d_out, int out_size, void* d_ws, size_t ws_size,
                              hipStream_t stream) {
    (void)in_sizes; (void)n_in; (void)out_size; (void)ws_size;
    const float* x      = (const float*)d_in[0];
    const float* conv_w = (const float*)d_in[1];
    const float* conv_b = (const float*)d_in[2];
    const float* Wl     = (const float*)d_in[3];
    const float* bl     = (const float*)d_in[4];
    const float* ln_g   = (const float*)d_in[5];
    const float* ln_b   = (const float*)d_in[6];
    const float* Win    = (const float*)d_in[7];
    const float* bi     = (const float*)d_in[8];
    const float* Wout   = (const float*)d_in[9];
    const float* bout   = (const float*)d_in[10];
    const float* W1     = (const float*)d_in[11];
    const float* b1     = (const float*)d_in[12];
    const float* W2     = (const float*)d_in[13];
    const float* b2     = (const float*)d_in[14];
    const float* g1     = (const float*)d_in[15];
    const float* be1    = (const float*)d_in[16];
    const float* g2     = (const float*)d_in[17];
    const float* be2    = (const float*)d_in[18];

    float* h   = (float*)d_ws;          // ROWS*4  floats
    float* qkv = h + ROWS * 4;          // ROWS*12 floats
    float* att = qkv + ROWS * 12;       // ROWS*4  floats  (~2 MB total)

    k_pre<<<BATCH, 128, 0, stream>>>(x, conv_w, conv_b, Wl, bl, ln_g, ln_b, h);
    for (int i = 0; i < 4; ++i) {
        k_qkv    <<<ROWS / 256, 256, 0, stream>>>(h, Win + i*48, bi + i*12, qkv);
        k_attn   <<<SEQ * 4,    256, 0, stream>>>(qkv, att);
        k_proj_ln<<<ROWS / 256, 256, 0, stream>>>(att, Wout + i*16, bout + i*4,
                                                  g1 + i*4, be1 + i*4, h);
        k_ffn    <<<ROWS / 64,  128, 0, stream>>>(h, W1 + i*FFDIM*4, b1 + i*FFDIM,
                                                  W2 + i*4*FFDIM, b2 + i*4,
                                                  g2 + i*4, be2 + i*4, h);
    }
    k_final<<<ROWS / 256, 256, 0, stream>>>(h, Wl, bl, (float*)d_out);
}